// ChemAttentionBlock_89206470738303
// MI455X (gfx1250) — compile-verified
//
#include <hip/hip_runtime.h>
#include <hip/hip_bf16.h>
#include <stdint.h>

typedef __attribute__((ext_vector_type(2))) float v2f;
typedef __attribute__((ext_vector_type(8))) float v8f;

#define NNODE   10000
#define NEDGE   160000
#define EN      (NEDGE + NNODE)        // edges + self loops
#define DIM     256
#define MT      (NNODE / 16)           // 625 row tiles
#define NEG_SLOPE 0.2f
#define BN_EPS  1e-5f

// ---------------------------------------------------------------------------
// 0. zero the accumulator span of the workspace
// ---------------------------------------------------------------------------
__global__ void zero_ws(float4* __restrict__ p, long n4) {
    long i = (long)blockIdx.x * blockDim.x + threadIdx.x;
    long stride = (long)gridDim.x * blockDim.x;
    float4 z = make_float4(0.f, 0.f, 0.f, 0.f);
    for (; i < n4; i += stride) p[i] = z;
}

// ---------------------------------------------------------------------------
// 1. dual GEMM: xl = x@Wl + bl, xr = x@Wr + br  via V_WMMA_F32_16X16X4_F32
//    one wave per (row-tile, col-tile, matrix); 625*16*2 = 20000 waves
// ---------------------------------------------------------------------------
__global__ void gemm_xlxr(const float* __restrict__ x,
                          const float* __restrict__ Wl, const float* __restrict__ bl,
                          const float* __restrict__ Wr, const float* __restrict__ br,
                          float* __restrict__ xl, float* __restrict__ xr) {
    int wid  = (int)((blockIdx.x * blockDim.x + threadIdx.x) >> 5);
    int lane = (int)(threadIdx.x & 31);
    if (wid >= MT * 32) return;

    int mt   = wid >> 5;          // 0..624
    int rest = wid & 31;
    int mat  = rest >> 4;         // 0 = left, 1 = right
    int nt   = rest & 15;         // 0..15 column tile

    const float* W    = mat ? Wr : Wl;
    const float* bias = mat ? br : bl;
    float*       out  = mat ? xr : xl;

    int half = lane >> 4;                 // 0 or 1
    int l16  = lane & 15;
    int row  = mt * 16 + l16;             // A fragment row (M)
    int col  = nt * 16 + l16;             // B/C fragment column (N)

    // A: lanes0-15 hold K=k+0,k+1; lanes16-31 hold K=k+2,k+3 (contiguous f32x2)
    const float* aptr = x + (long)row * DIM + 2 * half;
    // B: same K split; W row-major [K][N] so per-k step is DIM floats
    const float* bptr = W + (long)(2 * half) * DIM + col;

    v8f c = {};
#pragma unroll 4
    for (int k = 0; k < DIM; k += 4) {
        v2f a, b;
        a[0] = aptr[k];
        a[1] = aptr[k + 1];
        b[0] = bptr[(long)k * DIM];
        b[1] = bptr[(long)k * DIM + DIM];
        c = __builtin_amdgcn_wmma_f32_16x16x4_f32(
                /*neg_a=*/false, a, /*neg_b=*/false, b,
                /*c_mod=*/(short)0, c, /*reuse_a=*/false, /*reuse_b=*/false);
    }

    float bv = bias[col];
#pragma unroll
    for (int r = 0; r < 8; ++r) {
        int m = mt * 16 + r + 8 * half;   // C layout: VGPR r -> M=r (+8 for hi lanes)
        out[(long)m * DIM + col] = c[r] + bv;
    }
}

// ---------------------------------------------------------------------------
// 2. per-edge attention logits + segment max (order-preserving uint atomicMax)
// ---------------------------------------------------------------------------
__device__ __forceinline__ unsigned fenc(float f) {
    unsigned b = __float_as_uint(f);
    return (b & 0x80000000u) ? ~b : (b | 0x80000000u);
}
__device__ __forceinline__ float fdec(unsigned u) {
    return __uint_as_float((u & 0x80000000u) ? (u ^ 0x80000000u) : ~u);
}

__global__ void edge_logits(const int* __restrict__ ei,
                            const float* __restrict__ xl, const float* __restrict__ xr,
                            const float* __restrict__ att,
                            float* __restrict__ alpha_raw, unsigned* __restrict__ amax) {
    int wid  = (int)((blockIdx.x * blockDim.x + threadIdx.x) >> 5);
    int lane = (int)(threadIdx.x & 31);
    if (wid >= EN) return;

    int s, d;
    if (wid < NEDGE) { s = ei[wid]; d = ei[NEDGE + wid]; }
    else             { s = d = wid - NEDGE; }

    const float4* xs = (const float4*)(xl + (long)s * DIM);
    const float4* xd = (const float4*)(xr + (long)d * DIM);
    const float4* av = (const float4*)att;

    float acc = 0.f;
#pragma unroll
    for (int j = lane; j < DIM / 4; j += 32) {
        float4 a = xs[j], b = xd[j], t = av[j];
        float h;
        h = a.x + b.x; h = h > 0.f ? h : NEG_SLOPE * h; acc += t.x * h;
        h = a.y + b.y; h = h > 0.f ? h : NEG_SLOPE * h; acc += t.y * h;
        h = a.z + b.z; h = h > 0.f ? h : NEG_SLOPE * h; acc += t.z * h;
        h = a.w + b.w; h = h > 0.f ? h : NEG_SLOPE * h; acc += t.w * h;
    }
#pragma unroll
    for (int off = 16; off > 0; off >>= 1) acc += __shfl_xor(acc, off, 32);

    if (lane == 0) {
        alpha_raw[wid] = acc;
        atomicMax(&amax[d], fenc(acc));
    }
}

// ---------------------------------------------------------------------------
// 3. per-edge: alpha = exp(raw - amax[dst]); denom += alpha; agg[dst] += alpha*xl[src]
// ---------------------------------------------------------------------------
__global__ void edge_aggregate(const int* __restrict__ ei,
                               const float* __restrict__ xl,
                               const float* __restrict__ alpha_raw,
                               const unsigned* __restrict__ amax,
                               float* __restrict__ denom, float* __restrict__ agg) {
    int wid  = (int)((blockIdx.x * blockDim.x + threadIdx.x) >> 5);
    int lane = (int)(threadIdx.x & 31);
    if (wid >= EN) return;

    int s, d;
    if (wid < NEDGE) { s = ei[wid]; d = ei[NEDGE + wid]; }
    else             { s = d = wid - NEDGE; }

    float al = __expf(alpha_raw[wid] - fdec(amax[d]));
    if (lane == 0) atomicAdd(&denom[d], al);

    const float4* xs = (const float4*)(xl + (long)s * DIM);
    float* out = agg + (long)d * DIM;
#pragma unroll
    for (int j = lane; j < DIM / 4; j += 32) {
        float4 a = xs[j];
        atomicAdd(&out[4 * j + 0], al * a.x);
        atomicAdd(&out[4 * j + 1], al * a.y);
        atomicAdd(&out[4 * j + 2], al * a.z);
        atomicAdd(&out[4 * j + 3], al * a.w);
    }
}

// ---------------------------------------------------------------------------
// 4. normalize + bias + relu + dropout, accumulate per-column sum / sumsq
// ---------------------------------------------------------------------------
#define ROWS_PER_BLK 50

__global__ void post_stats(float* __restrict__ agg, const float* __restrict__ denom,
                           const float* __restrict__ bias, const float* __restrict__ du,
                           float* __restrict__ sums, float* __restrict__ sumsq) {
    int dcol = (int)threadIdx.x;           // 256 threads = 256 columns
    int i0   = (int)blockIdx.x * ROWS_PER_BLK;
    float s = 0.f, s2 = 0.f;
    for (int r = 0; r < ROWS_PER_BLK; ++r) {
        int i = i0 + r;
        if (i >= NNODE) break;
        long idx = (long)i * DIM + dcol;
        float v = agg[idx] / denom[i] + bias[dcol];
        v = v > 0.f ? v : 0.f;                              // relu
        v = (du[idx] >= 0.5f) ? 2.0f * v : 0.0f;            // dropout p=0.5
        agg[idx] = v;                                       // in-place
        s += v; s2 += v * v;
    }
    atomicAdd(&sums[dcol], s);
    atomicAdd(&sumsq[dcol], s2);
}

// ---------------------------------------------------------------------------
// 5. batchnorm finalize
// ---------------------------------------------------------------------------
__global__ void finalize(const float* __restrict__ v,
                         const float* __restrict__ sums, const float* __restrict__ sumsq,
                         const float* __restrict__ gamma, const float* __restrict__ beta,
                         float* __restrict__ out) {
    int dcol = (int)threadIdx.x;
    int i0   = (int)blockIdx.x * ROWS_PER_BLK;
    const float invn = 1.0f / (float)NNODE;
    float mean  = sums[dcol] * invn;
    float var   = sumsq[dcol] * invn - mean * mean;
    float scale = gamma[dcol] * rsqrtf(var + BN_EPS);
    float shift = beta[dcol] - mean * scale;
    for (int r = 0; r < ROWS_PER_BLK; ++r) {
        int i = i0 + r;
        if (i >= NNODE) break;
        long idx = (long)i * DIM + dcol;
        out[idx] = v[idx] * scale + shift;
    }
}

// ---------------------------------------------------------------------------
extern "C" void kernel_launch(void* const* d_in, const int* in_sizes, int n_in,
                              void* d_out, int out_size, void* d_ws, size_t ws_size,
                              hipStream_t stream) {
    const float* x     = (const float*)d_in[0];
    const int*   ei    = (const int*)d_in[1];
    const float* Wl    = (const float*)d_in[2];
    const float* bl    = (const float*)d_in[3];
    const float* Wr    = (const float*)d_in[4];
    const float* br    = (const float*)d_in[5];
    const float* att   = (const float*)d_in[6];
    const float* bias  = (const float*)d_in[7];
    const float* gamma = (const float*)d_in[8];
    const float* beta  = (const float*)d_in[9];
    const float* du    = (const float*)d_in[10];
    float*       out   = (float*)d_out;

    // workspace layout (floats); zeroed span first, then read-write temporaries
    float*    ws    = (float*)d_ws;
    float*    agg   = ws;                                   // N*DIM
    unsigned* amax  = (unsigned*)(ws + (long)NNODE * DIM);  // N
    float*    denom = ws + (long)NNODE * DIM + NNODE;       // N
    float*    sums  = denom + NNODE;                        // DIM
    float*    sumsq = sums + DIM;                           // DIM
    float*    xl    = sumsq + DIM;                          // N*DIM
    float*    xr    = xl + (long)NNODE * DIM;               // N*DIM
    float*    araw  = xr + (long)NNODE * DIM;               // EN

    // 0. zero [agg | amax | denom | sums | sumsq]
    long zeroN  = (long)NNODE * DIM + 2L * NNODE + 2L * DIM;
    long zeroN4 = zeroN / 4;   // all components are multiples of 4 floats
    zero_ws<<<1024, 256, 0, stream>>>((float4*)d_ws, zeroN4);

    // 1. dual GEMM: 625*32 waves, 4 waves/block
    gemm_xlxr<<<(MT * 32) / 4, 128, 0, stream>>>(x, Wl, bl, Wr, br, xl, xr);

    // 2. edge logits + segment max: one wave per edge, 8 waves/block
    int eblocks = (EN * 32 + 255) / 256;
    edge_logits<<<eblocks, 256, 0, stream>>>(ei, xl, xr, att, araw, amax);

    // 3. softmax numerator scatter-add
    edge_aggregate<<<eblocks, 256, 0, stream>>>(ei, xl, araw, amax, denom, agg);

    // 4. epilogue + column stats
    int sblocks = (NNODE + ROWS_PER_BLK - 1) / ROWS_PER_BLK;
    post_stats<<<sblocks, DIM, 0, stream>>>(agg, denom, bias, du, sums, sumsq);

    // 5. batchnorm
    finalize<<<sblocks, DIM, 0, stream>>>(agg, sums, sumsq, gamma, beta, out);
}